// RegionLoss_14250701488519
// MI455X (gfx1250) — compile-verified
//
#include <hip/hip_runtime.h>
#include <math.h>

// ---------------------------------------------------------------------------
// YOLOv2 RegionLoss, MI455X (gfx1250, wave32).
// Key optimization: only channels 0..4 of the 25 per-anchor channels are
// needed densely (18.5 MB streamed); the 20 class channels are gathered only
// at <=30 matched cells per batch. Cross-lane reduction uses
// V_WMMA_F32_16X16X4_F32 (ones-matrix trick) -> gfx1250 matrix path.
// ---------------------------------------------------------------------------

#define NB    128
#define NA    5
#define NC    20
#define NH    38
#define NW    38
#define NT    30
#define HW    (NH * NW)        // 1444
#define CH    (5 + NC)         // 25
#define CELLS (NA * HW)        // 7220 cells per batch
#define REC   16               // floats per ground-truth record
#define BLOCK 256
#define TILES ((CELLS + BLOCK - 1) / BLOCK)  // 29

typedef float v2f __attribute__((ext_vector_type(2)));
typedef float v8f __attribute__((ext_vector_type(8)));

__device__ __constant__ float c_aw[NA] = {1.3221f, 3.19275f, 5.05587f, 9.47112f, 11.2364f};
__device__ __constant__ float c_ah[NA] = {1.73145f, 4.00944f, 8.09892f, 4.84053f, 10.0071f};

__device__ __forceinline__ float sigm(float v) { return 1.f / (1.f + __expf(-v)); }

__device__ __forceinline__ float iou_cwh(float cx1, float cy1, float w1, float h1,
                                         float cx2, float cy2, float w2, float h2) {
    float l1 = cx1 - 0.5f * w1, r1 = cx1 + 0.5f * w1;
    float t1 = cy1 - 0.5f * h1, d1 = cy1 + 0.5f * h1;
    float l2 = cx2 - 0.5f * w2, r2 = cx2 + 0.5f * w2;
    float t2 = cy2 - 0.5f * h2, d2 = cy2 + 0.5f * h2;
    float cw = fminf(r1, r2) - fmaxf(l1, l2);
    float ch = fminf(d1, d2) - fmaxf(t1, t2);
    float inter = fmaxf(cw, 0.f) * fmaxf(ch, 0.f);
    return inter / (w1 * h1 + w2 * h2 - inter);
}

// One wave per batch: build 30 ground-truth records.
// rec layout (16 f32): [0]=valid, [1]=cell(int), [2]=cls(int), [3]=coord_mask,
// [4]=tx, [5]=ty, [6]=tw, [7]=th, [8]=tconf, [9..12]=gx,gy,gw,gh (scaled).
__global__ void region_prep(const float* __restrict__ out,
                            const float* __restrict__ tgt,
                            float* __restrict__ rec,
                            float* __restrict__ loss_out) {
    int b = blockIdx.x;
    int t = threadIdx.x;
    if (b == 0 && t == 0) loss_out[0] = 0.f;   // stream-ordered before main pass
    if (t >= NT) return;

    const float* tr = tgt + (b * NT + t) * 5;
    float clsf = tr[0];
    float gx = tr[1] * (float)NW, gy = tr[2] * (float)NH;
    float gw = tr[3] * (float)NW, gh = tr[4] * (float)NH;

    // best anchor: argmax of shape IoU (boxes at origin), first max wins
    int bn = 0; float best = -1.f;
#pragma unroll
    for (int a = 0; a < NA; ++a) {
        float aw = c_aw[a], ah = c_ah[a];
        float inter = fminf(gw, aw) * fminf(gh, ah);
        float uni = gw * gh + aw * ah - inter;
        float ratio = inter / uni;
        if (ratio > best) { best = ratio; bn = a; }
    }
    int gi = (int)gx; gi = min(max(gi, 0), NW - 1);
    int gj = (int)gy; gj = min(max(gj, 0), NH - 1);

    // gather pred box at (b, bn, gj, gi)
    int base = ((b * NA + bn) * CH) * HW + gj * NW + gi;
    float o0 = out[base], o1 = out[base + HW];
    float o2 = out[base + 2 * HW], o3 = out[base + 3 * HW];
    float aw = c_aw[bn], ah = c_ah[bn];
    float px = sigm(o0) + (float)gi;
    float py = sigm(o1) + (float)gj;
    float pw = __expf(o2) * aw;
    float ph = __expf(o3) * ah;
    float iou_t = iou_cwh(px, py, pw, ph, gx, gy, gw, gh);

    float* r = rec + (b * NT + t) * REC;
    r[0]  = (clsf >= 0.f) ? 1.f : 0.f;
    r[1]  = __int_as_float((bn * NH + gj) * NW + gi);
    r[2]  = __int_as_float((int)clsf);
    r[3]  = 2.f - tr[3] * tr[4];      // COORD_SCALE * (2 - gw_frac*gh_frac)
    r[4]  = gx - (float)gi;
    r[5]  = gy - (float)gj;
    r[6]  = __logf(gw / aw);
    r[7]  = __logf(gh / ah);
    r[8]  = iou_t;
    r[9]  = gx; r[10] = gy; r[11] = gw; r[12] = gh;
    r[13] = 0.f; r[14] = 0.f; r[15] = 0.f;
}

// Streaming pass over channels 0..4 of every cell.
__global__ void region_main(const float* __restrict__ out,
                            const float* __restrict__ rec,
                            float* __restrict__ loss_out) {
    __shared__ float sg[NT * REC];

    int b    = blockIdx.x / TILES;
    int tile = blockIdx.x % TILES;
    int tid  = threadIdx.x;

    for (int k = tid; k < NT * REC; k += BLOCK)
        sg[k] = rec[b * NT * REC + k];
    __syncthreads();

    float partial = 0.f;
    int cell = tile * BLOCK + tid;
    if (cell < CELLS) {
        int a   = cell / HW;
        int rem = cell - a * HW;
        int j   = rem / NW;
        int i   = rem - j * NW;
        int base = ((b * NA + a) * CH) * HW + rem;   // channel stride = HW

        float o0 = out[base],          o1 = out[base + HW];
        float o2 = out[base + 2 * HW], o3 = out[base + 3 * HW];
        float o4 = out[base + 4 * HW];

        float x = sigm(o0), y = sigm(o1);
        float wv = o2, hv = o3;
        float conf = sigm(o4);
        float aw = c_aw[a], ah = c_ah[a];
        float px = x + (float)i, py = y + (float)j;
        float pw = __expf(wv) * aw, ph = __expf(hv) * ah;

        float maxiou = 0.f;
        int match = -1;                  // last valid gt targeting this cell wins
        for (int t = 0; t < NT; ++t) {
            const float* r = sg + t * REC;
            if (__float_as_int(r[0]) != 0) {
                float iou = iou_cwh(px, py, pw, ph, r[9], r[10], r[11], r[12]);
                maxiou = fmaxf(maxiou, iou);
                if (__float_as_int(r[1]) == cell) match = t;
            }
        }

        if (match < 0) {
            // defaults: coord_mask=0.01, tx=ty=0.5, tw=th=0, tconf=0,
            // conf_mask = (max_iou > 0.6) ? 0 : NOOBJECT_SCALE(1)
            float noobj = (maxiou > 0.6f) ? 0.f : 1.f;
            float dx = x - 0.5f, dy = y - 0.5f;
            partial = 0.01f * (dx * dx + dy * dy + wv * wv + hv * hv)
                    + noobj * conf * conf;
        } else {
            const float* r = sg + match * REC;
            float cm = r[3];
            float dx = x - r[4], dy = y - r[5];
            float dw = wv - r[6], dh = hv - r[7];
            float dc = conf - r[8];
            partial = cm * (dx * dx + dy * dy + dw * dw + dh * dh)
                    + 5.f * dc * dc;                      // OBJECT_SCALE
            // class loss: -log_softmax picked at target class (rare path)
            int cls = __float_as_int(r[2]);
            const float* lp = out + base + 5 * HW;
            float l[NC];
            float m = -INFINITY;
#pragma unroll
            for (int c = 0; c < NC; ++c) { l[c] = lp[c * HW]; m = fmaxf(m, l[c]); }
            float s = 0.f;
#pragma unroll
            for (int c = 0; c < NC; ++c) s += __expf(l[c] - m);
            float lse = m + __logf(s);
            partial += lse - l[cls];                      // CLASS_SCALE * -picked
        }
    }

    // ------------------------------------------------------------------
    // Wave-level reduction via V_WMMA_F32_16X16X4_F32 (EXEC all-1s: every
    // wave of the full 256-thread block executes this unconditionally).
    // A (16x4 f32, 2 VGPRs/lane): lanes 0-15 hold K=0,1 for M=lane;
    // lanes 16-31 hold K=2,3 for M=lane-16. Put partial in K0/K2, 0 in K1/K3.
    // With B = ones: D[m][n] = partial[m] + partial[m+16] for every n.
    // Sum the 8 D VGPRs (M=0..7 on lanes 0-15, M=8..15 on lanes 16-31),
    // fold with shfl_xor(16) -> full 32-lane sum in every lane.
    // ------------------------------------------------------------------
    v2f A; A.x = partial; A.y = 0.f;
    v2f B; B.x = 1.f;     B.y = 1.f;
    v8f C = {};
    v8f D = __builtin_amdgcn_wmma_f32_16x16x4_f32(
        /*neg_a=*/false, A, /*neg_b=*/false, B,
        /*c_mod=*/(short)0, C, /*reuse_a=*/false, /*reuse_b=*/false);
    float wsum = D[0] + D[1] + D[2] + D[3] + D[4] + D[5] + D[6] + D[7];
    wsum += __shfl_xor(wsum, 16, 32);

    if ((tid & 31) == 0)
        atomicAdd(loss_out, wsum);      // global_atomic_add_f32, 8 per block
}

extern "C" void kernel_launch(void* const* d_in, const int* in_sizes, int n_in,
                              void* d_out, int out_size, void* d_ws, size_t ws_size,
                              hipStream_t stream) {
    const float* out = (const float*)d_in[0];   // (128, 125, 38, 38) f32
    const float* tgt = (const float*)d_in[1];   // (128, 150) f32
    float* loss = (float*)d_out;                // scalar f32
    float* rec  = (float*)d_ws;                 // 128*30*16 f32 = 240 KB scratch

    region_prep<<<NB, 32, 0, stream>>>(out, tgt, rec, loss);
    region_main<<<NB * TILES, BLOCK, 0, stream>>>(out, rec, loss);
}